// VectorQuantizer_40845138985506
// MI455X (gfx1250) — compile-verified
//
#include <hip/hip_runtime.h>

typedef float v2f __attribute__((ext_vector_type(2)));
typedef float v8f __attribute__((ext_vector_type(8)));
typedef int   v4i __attribute__((ext_vector_type(4)));

#define C_DIM 512
#define NB    2048     // number of codes
#define BATCH 1024
#define NTOK  64

#if __has_builtin(__builtin_amdgcn_global_load_async_to_lds_b128) && \
    __has_builtin(__builtin_amdgcn_s_wait_asynccnt)
#define USE_ASYNC_LDS 1
#else
#define USE_ASYNC_LDS 0
#endif

__device__ __forceinline__ float invw(int c) {
  // BOUNDS = (0,256,384,448,512) -> group widths 256,128,64,64
  if (c < 256) return 1.0f / 256.0f;
  if (c < 384) return 1.0f / 128.0f;
  return 1.0f / 64.0f;
}

// ---------------------------------------------------------------------------
// Kernel 0: emb = base @ proj_W^T  via V_WMMA_F32_16X16X4_F32
// grid: 1024 blocks x 128 threads (4 waves); each wave one 16x16 tile
// ---------------------------------------------------------------------------
__global__ void k_embgemm(const float* __restrict__ base,
                          const float* __restrict__ projW,
                          float* __restrict__ emb) {
  int wave = threadIdx.x >> 5;
  int lane = threadIdx.x & 31;
  int t    = blockIdx.x * 4 + wave;        // 4096 tiles = 128 x 32
  int m0   = (t >> 5) * 16;
  int n0   = (t & 31) * 16;
  int l15  = lane & 15;
  int lh   = lane >> 4;

  const float* ap = base  + (size_t)(m0 + l15) * C_DIM + 2 * lh;
  const float* bp = projW + (size_t)(n0 + l15) * C_DIM + 2 * lh;

  v8f acc = {0.f, 0.f, 0.f, 0.f, 0.f, 0.f, 0.f, 0.f};
#pragma unroll 8
  for (int k0 = 0; k0 < C_DIM; k0 += 4) {
    v2f a = *(const v2f*)(ap + k0);
    v2f b = *(const v2f*)(bp + k0);
    acc = __builtin_amdgcn_wmma_f32_16x16x4_f32(false, a, false, b,
                                                (short)0, acc, false, false);
  }
#pragma unroll
  for (int r = 0; r < 8; ++r)
    emb[(size_t)(m0 + r + 8 * lh) * C_DIM + n0 + l15] = acc[r];
}

// ---------------------------------------------------------------------------
// bias[j] = sum_c emb[j][c]^2 / w(c)   (2048 blocks x 128 threads)
// ---------------------------------------------------------------------------
__global__ void k_bias(const float* __restrict__ emb, float* __restrict__ bias) {
  __shared__ float red[4];
  int j = blockIdx.x;
  int c = threadIdx.x * 4;
  float4 e = *(const float4*)(emb + (size_t)j * C_DIM + c);
  float v = e.x * e.x * invw(c) + e.y * e.y * invw(c + 1) +
            e.z * e.z * invw(c + 2) + e.w * e.w * invw(c + 3);
  for (int off = 16; off; off >>= 1) v += __shfl_down(v, off, 32);
  int lane = threadIdx.x & 31, w = threadIdx.x >> 5;
  if (lane == 0) red[w] = v;
  __syncthreads();
  if (threadIdx.x == 0) bias[j] = red[0] + red[1] + red[2] + red[3];
}

// ---------------------------------------------------------------------------
// Pack scaled E' as K-pairs: embB[p][j] = { emb[j][2p]*invw, emb[j][2p+1]*invw }
// so a WMMA B-fragment lane load is one contiguous b64. (2048 blocks x 256)
// ---------------------------------------------------------------------------
__global__ void k_pack(const float* __restrict__ emb, float* __restrict__ embB) {
  int g = blockIdx.x * 256 + threadIdx.x;   // 256*2048 entries
  int p = g >> 11;                          // K-pair index 0..255
  int j = g & (NB - 1);
  float a = emb[(size_t)j * C_DIM + 2 * p]     * invw(2 * p);
  float b = emb[(size_t)j * C_DIM + 2 * p + 1] * invw(2 * p + 1);
  float2* dst = (float2*)embB;
  dst[(size_t)p * NB + j] = make_float2(a, b);
}

// ---------------------------------------------------------------------------
// Area downsample: rest[r=b*pn+p][c] = mean over g=64/pn rows of src (B,N,C)
// ---------------------------------------------------------------------------
__global__ void k_down(const float* __restrict__ src, float* __restrict__ rest,
                       int pn) {
  int r = blockIdx.x;
  int b = r / pn, p = r % pn;
  int g = NTOK / pn;
  int c = threadIdx.x * 4;
  float sx = 0.f, sy = 0.f, sz = 0.f, sw = 0.f;
  const float* base = src + ((size_t)b * NTOK + (size_t)p * g) * C_DIM + c;
  for (int jj = 0; jj < g; ++jj) {
    float4 v = *(const float4*)(base + (size_t)jj * C_DIM);
    sx += v.x; sy += v.y; sz += v.z; sw += v.w;
  }
  float inv = 1.0f / (float)g;
  float4 o = make_float4(sx * inv, sy * inv, sz * inv, sw * inv);
  *(float4*)(rest + (size_t)r * C_DIM + c) = o;
}

// ---------------------------------------------------------------------------
// Fused distance GEMM + argmin.
// Block: 256 threads = 8 waves, 32 query rows staged in LDS (64 KB, async).
// Wave w covers codes [w*256, w*256+256) as 2 chunks x 8 16-wide tiles;
// each B-fragment feeds TWO wmma (row tiles 0-15 and 16-31) for 2x B reuse:
// per K-step: 2 ds_load_b64 + 8 global_load_b64 + 16 v_wmma_f32_16x16x4_f32.
// Argmin reduction reuses the query-tile LDS (dead after the K-loops).
// ---------------------------------------------------------------------------
__global__ void __launch_bounds__(256)
k_dist(const float* __restrict__ q, const float* __restrict__ embB,
       const float* __restrict__ bias, int* __restrict__ idxOut) {
  __shared__ float qt[32 * C_DIM];        // 64 KB query tile (2 row-tiles)

  int tid  = threadIdx.x;
  int row0 = blockIdx.x * 32;

  // stage 32x512 query tile: 4096 float4 / 256 threads = 16 each
  {
    const float4* qg = (const float4*)(q + (size_t)row0 * C_DIM);
    float4* qs = (float4*)qt;
#if USE_ASYNC_LDS
#pragma unroll
    for (int i = 0; i < 16; ++i) {
      v4i* gp = (v4i*)(qg + tid + 256 * i);   // drop const, generic AS
      v4i* lp = (v4i*)(qs + tid + 256 * i);
      __builtin_amdgcn_global_load_async_to_lds_b128(
          (__attribute__((address_space(1))) v4i*)gp,
          (__attribute__((address_space(3))) v4i*)lp,
          0, 0);
    }
    __builtin_amdgcn_s_wait_asynccnt(0);
#else
#pragma unroll
    for (int i = 0; i < 16; ++i) qs[tid + 256 * i] = qg[tid + 256 * i];
#endif
  }
  __syncthreads();

  int wave = tid >> 5, lane = tid & 31;
  int l15 = lane & 15, lh = lane >> 4;
  const float* a0row = qt + (size_t)l15 * C_DIM + 2 * lh;         // rows 0-15
  const float* a1row = qt + (size_t)(16 + l15) * C_DIM + 2 * lh;  // rows 16-31

  float minv[16];
  int   mini[16];
#pragma unroll
  for (int r = 0; r < 16; ++r) { minv[r] = 3.4e38f; mini[r] = 0; }

  for (int chunk = 0; chunk < 2; ++chunk) {
    int cb = wave * 256 + chunk * 128;    // 8 tiles of 16 codes
    v8f acc[16];
#pragma unroll
    for (int t = 0; t < 16; ++t)
      acc[t] = (v8f){0.f, 0.f, 0.f, 0.f, 0.f, 0.f, 0.f, 0.f};

    for (int k0 = 0; k0 < C_DIM; k0 += 4) {
      v2f a0 = *(const v2f*)(a0row + k0);   // ds_load_b64 from LDS
      v2f a1 = *(const v2f*)(a1row + k0);
      const float* bbase = embB + ((size_t)((k0 >> 1) + lh)) * (2 * NB)
                         + 2 * (cb + l15);
#pragma unroll
      for (int t = 0; t < 8; ++t) {
        v2f b = *(const v2f*)(bbase + t * 32);   // global b64, L2-resident
        acc[t] = __builtin_amdgcn_wmma_f32_16x16x4_f32(false, a0, false, b,
                                                       (short)0, acc[t],
                                                       false, false);
        acc[8 + t] = __builtin_amdgcn_wmma_f32_16x16x4_f32(false, a1, false, b,
                                                           (short)0, acc[8 + t],
                                                           false, false);
      }
    }
    // fold into running per-lane argmin (cols ascend -> '<' keeps lowest idx)
#pragma unroll
    for (int t = 0; t < 8; ++t) {
      int col = cb + t * 16 + l15;
      float bj = bias[col];
#pragma unroll
      for (int r = 0; r < 8; ++r) {
        float d0 = bj - 2.0f * acc[t][r];
        if (d0 < minv[r]) { minv[r] = d0; mini[r] = col; }
        float d1 = bj - 2.0f * acc[8 + t][r];
        if (d1 < minv[8 + r]) { minv[8 + r] = d1; mini[8 + r] = col; }
      }
    }
  }

  // reduce over the 16 lanes of each half (rows M = r + 8*lh live there)
#pragma unroll
  for (int m = 1; m < 16; m <<= 1) {
#pragma unroll
    for (int r = 0; r < 16; ++r) {
      float ov = __shfl_xor(minv[r], m, 32);
      int   oi = __shfl_xor(mini[r], m, 32);
      if (ov < minv[r] || (ov == minv[r] && oi < mini[r])) {
        minv[r] = ov; mini[r] = oi;
      }
    }
  }

  // query tile is dead now: reuse its LDS for the cross-wave reduction
  __syncthreads();
  float* redV = qt;                 // [32][8]
  int*   redI = (int*)(qt + 256);   // [32][8]
  if (l15 == 0) {
#pragma unroll
    for (int r = 0; r < 8; ++r) {
      int row = lh * 8 + r;
      redV[row * 8 + wave] = minv[r];
      redI[row * 8 + wave] = mini[r];
      redV[(16 + row) * 8 + wave] = minv[8 + r];
      redI[(16 + row) * 8 + wave] = mini[8 + r];
    }
  }
  __syncthreads();
  if (tid < 32) {
    float bv = redV[tid * 8];
    int   bi = redI[tid * 8];
#pragma unroll
    for (int w = 1; w < 8; ++w) {
      float v = redV[tid * 8 + w];
      int  i2 = redI[tid * 8 + w];
      if (v < bv || (v == bv && i2 < bi)) { bv = v; bi = i2; }
    }
    idxOut[row0 + tid] = bi;
  }
}

// ---------------------------------------------------------------------------
// Fused gather + linear upsample + f_hat/f_rest update + elem_dist accumulate.
// grid: B*N = 65536 blocks x 128 threads
// ---------------------------------------------------------------------------
__global__ void k_update(const float* __restrict__ f, const float* __restrict__ emb,
                         const int* __restrict__ idx, float* __restrict__ fhat,
                         float* __restrict__ frest, float* __restrict__ qacc,
                         int pn, int first, int last) {
  __shared__ float red[4];
  int bn = blockIdx.x;
  int b = bn >> 6, n = bn & 63;

  float src = (n + 0.5f) * ((float)pn / 64.0f) - 0.5f;
  src = fminf(fmaxf(src, 0.0f), (float)(pn - 1));
  int i0 = (int)floorf(src);
  int i1 = min(i0 + 1, pn - 1);
  float w = src - (float)i0;

  int j0 = idx[b * pn + i0];
  int j1 = idx[b * pn + i1];

  int c = threadIdx.x * 4;
  size_t off = (size_t)bn * C_DIM + c;
  float4 e0 = *(const float4*)(emb + (size_t)j0 * C_DIM + c);
  float4 e1 = *(const float4*)(emb + (size_t)j1 * C_DIM + c);
  float hx = e0.x * (1.f - w) + e1.x * w;
  float hy = e0.y * (1.f - w) + e1.y * w;
  float hz = e0.z * (1.f - w) + e1.z * w;
  float hw = e0.w * (1.f - w) + e1.w * w;

  float4 fv = *(const float4*)(f + off);
  float4 fh;
  if (first) { fh = make_float4(hx, hy, hz, hw); }
  else {
    float4 p = *(const float4*)(fhat + off);
    fh = make_float4(p.x + hx, p.y + hy, p.z + hz, p.w + hw);
  }
  *(float4*)(fhat + off) = fh;

  if (!last) {
    float4 ro;
    if (first) ro = make_float4(fv.x - hx, fv.y - hy, fv.z - hz, fv.w - hw);
    else {
      float4 p = *(const float4*)(frest + off);
      ro = make_float4(p.x - hx, p.y - hy, p.z - hz, p.w - hw);
    }
    *(float4*)(frest + off) = ro;
  }

  float dx = fh.x - fv.x, dy = fh.y - fv.y, dz = fh.z - fv.z, dw = fh.w - fv.w;
  float v = dx * dx * invw(c) + dy * dy * invw(c + 1) +
            dz * dz * invw(c + 2) + dw * dw * invw(c + 3);
  for (int o = 16; o; o >>= 1) v += __shfl_down(v, o, 32);
  int lane = threadIdx.x & 31, wv = threadIdx.x >> 5;
  if (lane == 0) red[wv] = v;
  __syncthreads();
  if (threadIdx.x == 0) {
    float s = red[0] + red[1] + red[2] + red[3];
    qacc[bn] = (first ? 0.f : qacc[bn]) + s;
  }
}

// commit = qacc*0.25/7, qlat = qacc/7
__global__ void k_final(const float* __restrict__ qacc,
                        float* __restrict__ commit, float* __restrict__ qlat) {
  int i = blockIdx.x * 256 + threadIdx.x;
  float a = qacc[i];
  qlat[i]   = a * (1.0f / 7.0f);
  commit[i] = a * (0.25f / 7.0f);
}

// ---------------------------------------------------------------------------
extern "C" void kernel_launch(void* const* d_in, const int* in_sizes, int n_in,
                              void* d_out, int out_size, void* d_ws, size_t ws_size,
                              hipStream_t stream) {
  const float* f     = (const float*)d_in[0];  // (1024, 64, 512)
  const float* base  = (const float*)d_in[1];  // (2048, 512)
  const float* projW = (const float*)d_in[2];  // (512, 512)

  float* fhat   = (float*)d_out;                                   // (B,N,C)
  float* commit = fhat + (size_t)BATCH * NTOK * C_DIM;             // (B,N)
  float* qlat   = commit + (size_t)BATCH * NTOK;                   // (B,N)

  char* ws = (char*)d_ws;
  float* emb   = (float*)ws;   ws += (size_t)NB * C_DIM * 4;            // 4 MB
  float* embB  = (float*)ws;   ws += (size_t)NB * C_DIM * 4;            // 4 MB
  float* bias  = (float*)ws;   ws += (size_t)NB * 4 + 1024;             // 8 KB
  float* frest = (float*)ws;   ws += (size_t)BATCH * NTOK * C_DIM * 4;  // 128 MB
  float* rest  = (float*)ws;   ws += (size_t)BATCH * 32 * C_DIM * 4;    // 64 MB
  int*   idxb  = (int*)ws;     ws += (size_t)BATCH * NTOK * 4;          // 256 KB
  float* qacc  = (float*)ws;

  k_embgemm<<<1024, 128, 0, stream>>>(base, projW, emb);
  k_bias<<<NB, 128, 0, stream>>>(emb, bias);
  k_pack<<<2048, 256, 0, stream>>>(emb, embB);

  static const int scales[7] = {1, 2, 4, 8, 16, 32, 64};
  for (int s = 0; s < 7; ++s) {
    int pn = scales[s];
    int R = BATCH * pn;
    int first = (s == 0), last = (s == 6);
    const float* qptr;
    if (pn < NTOK) {
      k_down<<<R, 128, 0, stream>>>(first ? f : frest, rest, pn);
      qptr = rest;
    } else {
      qptr = frest;   // g==1: residual (B,N,C) rows are exactly the queries
    }
    k_dist<<<R / 32, 256, 0, stream>>>(qptr, embB, bias, idxb);
    k_update<<<BATCH * NTOK, 128, 0, stream>>>(f, emb, idxb, fhat, frest, qacc,
                                               pn, first, last);
  }
  k_final<<<BATCH * NTOK / 256, 256, 0, stream>>>(qacc, commit, qlat);
}